// RecurrentGCN_25778393710731
// MI455X (gfx1250) — compile-verified
//
#include <hip/hip_runtime.h>

typedef __attribute__((ext_vector_type(2))) float v2f;
typedef __attribute__((ext_vector_type(8))) float v8f;

#define N_NODES_C 500000
#define N_FEAT_C  128
#define HIDDEN_C  32
#define CATDIM_C  160
#define LDS_K     132   // padded K stride: banks (4n+k)%64 distinct across the wave

__global__ __launch_bounds__(256) void rgcn_fused_wmma(
    const float* __restrict__ x,      // [N, 128]
    const float* __restrict__ w_z,    // [2,1,160,32]
    const float* __restrict__ w_h,    // [2,1,160,32]
    const float* __restrict__ b_z,    // [32]
    const float* __restrict__ b_h,    // [32]
    const float* __restrict__ w_lin,  // [32,1]
    const float* __restrict__ b_lin,  // [1]
    float* __restrict__ out)          // [N,1]
{
    __shared__ float ldsW[2][HIDDEN_C][LDS_K];

    // Stage effective weights transposed: ldsW[m][n][k] = w[0,0,k,n] + w[1,0,k,n]
    // (h-columns k>=128 of the weights multiply h0==0 and are dropped)
    for (int idx = threadIdx.x; idx < 2 * HIDDEN_C * N_FEAT_C; idx += 256) {
        int m   = idx >> 12;        // 4096 elements per matrix
        int rem = idx & 4095;
        int n   = rem & 31;
        int k   = rem >> 5;
        const float* w = (m == 0) ? w_z : w_h;
        ldsW[m][n][k] = w[k * 32 + n] + w[CATDIM_C * 32 + k * 32 + n];
    }
    __syncthreads();

    const int lane = threadIdx.x & 31;
    const int wave = threadIdx.x >> 5;
    const int tile = blockIdx.x * 8 + wave;     // one 16-row tile per wave
    const int rowBase = tile * 16;
    if (rowBase >= N_NODES_C) return;           // wave-uniform: EXEC stays all-1s

    const int half = lane >> 4;                 // 0: K={0,1}, 1: K={2,3} of each step
    const int l16  = lane & 15;
    const int kk   = half * 2;

    const float* xrow = x + (size_t)(rowBase + l16) * N_FEAT_C + kk;
    const float* wzp0 = &ldsW[0][l16][kk];
    const float* wzp1 = &ldsW[0][l16 + 16][kk];
    const float* whp0 = &ldsW[1][l16][kk];
    const float* whp1 = &ldsW[1][l16 + 16][kk];

    v8f cz0 = {}, cz1 = {}, ch0 = {}, ch1 = {};

    #pragma unroll 8
    for (int k0 = 0; k0 < N_FEAT_C; k0 += 4) {
        v2f a   = *(const v2f*)(xrow + k0);     // A fragment: 16x4 f32
        v2f bz0 = *(const v2f*)(wzp0 + k0);     // B fragments: 4x16 f32 from LDS
        v2f bz1 = *(const v2f*)(wzp1 + k0);
        v2f bh0 = *(const v2f*)(whp0 + k0);
        v2f bh1 = *(const v2f*)(whp1 + k0);
        cz0 = __builtin_amdgcn_wmma_f32_16x16x4_f32(false, a, false, bz0, (short)0, cz0, false, false);
        cz1 = __builtin_amdgcn_wmma_f32_16x16x4_f32(false, a, false, bz1, (short)0, cz1, false, false);
        ch0 = __builtin_amdgcn_wmma_f32_16x16x4_f32(false, a, false, bh0, (short)0, ch0, false, false);
        ch1 = __builtin_amdgcn_wmma_f32_16x16x4_f32(false, a, false, bh1, (short)0, ch1, false, false);
    }

    // Fused epilogue: z=sigmoid, h~=tanh, h=relu((1-z)*h~), out = h . w_lin + b_lin
    const int n0 = l16, n1 = l16 + 16;
    const float bz0v = b_z[n0], bz1v = b_z[n1];
    const float bh0v = b_h[n0], bh1v = b_h[n1];
    const float wl0  = w_lin[n0], wl1 = w_lin[n1];
    const float bl   = b_lin[0];

    #pragma unroll
    for (int i = 0; i < 8; ++i) {
        // C layout: VGPR i holds row M=i (lanes 0-15) / M=i+8 (lanes 16-31), N = l16 (+16 for *1 accs)
        float z0 = 1.0f / (1.0f + __expf(-(cz0[i] + bz0v)));
        float t0 = tanhf(ch0[i] + bh0v);
        float h0 = fmaxf((1.0f - z0) * t0, 0.0f);
        float z1 = 1.0f / (1.0f + __expf(-(cz1[i] + bz1v)));
        float t1 = tanhf(ch1[i] + bh1v);
        float h1 = fmaxf((1.0f - z1) * t1, 0.0f);
        float p  = h0 * wl0 + h1 * wl1;
        // half-wave reduction over 16 lanes (masks < 16 stay within each half)
        p += __shfl_xor(p, 1, 32);
        p += __shfl_xor(p, 2, 32);
        p += __shfl_xor(p, 4, 32);
        p += __shfl_xor(p, 8, 32);
        if (l16 == 0) {
            out[rowBase + i + (half << 3)] = p + bl;
        }
    }
}

extern "C" void kernel_launch(void* const* d_in, const int* in_sizes, int n_in,
                              void* d_out, int out_size, void* d_ws, size_t ws_size,
                              hipStream_t stream) {
    // setup_inputs order:
    // 0:x 1:edge_index(unused) 2:w_z 3:b_z 4:w_r(unused) 5:b_r(unused)
    // 6:w_h 7:b_h 8:w_lin 9:b_lin
    const float* x     = (const float*)d_in[0];
    const float* w_z   = (const float*)d_in[2];
    const float* b_z   = (const float*)d_in[3];
    const float* w_h   = (const float*)d_in[6];
    const float* b_h   = (const float*)d_in[7];
    const float* w_lin = (const float*)d_in[8];
    const float* b_lin = (const float*)d_in[9];
    float* out = (float*)d_out;

    int tiles  = (N_NODES_C + 15) / 16;   // 31250
    int blocks = (tiles + 7) / 8;         // 3907
    hipLaunchKernelGGL(rgcn_fused_wmma, dim3(blocks), dim3(256), 0, stream,
                       x, w_z, w_h, b_z, b_h, w_lin, b_lin, out);
}